// GumbelK_27599459844820
// MI455X (gfx1250) — compile-verified
//
#include <hip/hip_runtime.h>
#include <hip/hip_bf16.h>

typedef float v2f __attribute__((ext_vector_type(2)));
typedef float v8f __attribute__((ext_vector_type(8)));

#define DIMN   4096
#define KITER  32
#define TPB    256
#define EPT    (DIMN / TPB)           // 16 elements per thread
#define NWAVES (TPB / 32)             // 8 waves per block
#define EPSF   1.175494350822287508e-38f   // np.float32 tiny
#define LN2F   0.693147180559945309417f
#define NLN2F  (-0.693147180559945309417f)
#define C100L2E 144.269504088896340736f    // 100 / ln(2) : folds 1/t into exp2
#define NEGINF (-3.402823466e+38f)

// fast natural log: v_log_f32 (log2) * ln2 ; valid for normal positive args
__device__ __forceinline__ float fast_log(float x) {
    return __builtin_amdgcn_logf(x) * LN2F;
}

// ---- wave32 max reduction (shfl_xor tree) ----
__device__ __forceinline__ float wave_max(float v) {
#pragma unroll
    for (int off = 16; off > 0; off >>= 1)
        v = fmaxf(v, __shfl_xor(v, off, 32));
    return v;
}

// ---- wave32 sum reduction via V_WMMA_F32_16X16X4_F32 ----
// A(16x4): lane L (0-15): A[L][0]=p, A[L][1]=0 ; lane L (16-31): A[L-16][2]=p, A[L-16][3]=0
// B(4x16) = ones  ->  D[m,n] = p_m + p_{m+16}
// D layout: vgpr v, lanes 0-15 hold s_v, lanes 16-31 hold s_{v+8}.
// sum of 8 D vgprs then xor-16 shuffle -> full 32-lane sum in every lane.
__device__ __forceinline__ float wave_sum_wmma(float p) {
    v2f a; a.x = p;    a.y = 0.0f;
    v2f b; b.x = 1.0f; b.y = 1.0f;
    v8f c = {};
    v8f d = __builtin_amdgcn_wmma_f32_16x16x4_f32(
        /*neg_a=*/false, a, /*neg_b=*/false, b,
        /*c_mod=*/(short)0, c, /*reuse_a=*/false, /*reuse_b=*/false);
    float t = ((d[0] + d[1]) + (d[2] + d[3])) + ((d[4] + d[5]) + (d[6] + d[7]));
    t += __shfl_xor(t, 16, 32);
    return t;
}

// ---- block reductions across NWAVES waves via small LDS array ----
__device__ __forceinline__ float block_max(float v, float* s_red) {
    v = wave_max(v);
    const int lane = threadIdx.x & 31, wid = threadIdx.x >> 5;
    if (lane == 0) s_red[wid] = v;
    __syncthreads();
    float m = s_red[0];
#pragma unroll
    for (int w = 1; w < NWAVES; ++w) m = fmaxf(m, s_red[w]);
    __syncthreads();
    return m;
}

__device__ __forceinline__ float block_sum(float v, float* s_red) {
    v = wave_sum_wmma(v);
    const int lane = threadIdx.x & 31, wid = threadIdx.x >> 5;
    if (lane == 0) s_red[wid] = v;
    __syncthreads();
    float s = 0.0f;
#pragma unroll
    for (int w = 0; w < NWAVES; ++w) s += s_red[w];
    __syncthreads();
    return s;
}

// ---- prep: log_proba = log_softmax(w)  (one block, cold) ----
__global__ void __launch_bounds__(TPB)
gumbel_prep_logsoftmax(const float* __restrict__ w, float* __restrict__ lp) {
    __shared__ float s_red[NWAVES];
    const int t = threadIdx.x;
    float x[EPT];
    const float4* w4 = (const float4*)(w + t * EPT);
    float lm = NEGINF;
#pragma unroll
    for (int j = 0; j < EPT / 4; ++j) {
        float4 v = w4[j];
        x[4*j+0] = v.x; x[4*j+1] = v.y; x[4*j+2] = v.z; x[4*j+3] = v.w;
        lm = fmaxf(lm, fmaxf(fmaxf(v.x, v.y), fmaxf(v.z, v.w)));
    }
    const float m = block_max(lm, s_red);
    float ls = 0.0f;
#pragma unroll
    for (int i = 0; i < EPT; ++i)
        ls += __builtin_amdgcn_exp2f((x[i] - m) * 1.44269504088896340736f);
    const float s = block_sum(ls, s_red);
    const float lse = m + fast_log(s);
    float4* lp4 = (float4*)(lp + t * EPT);
#pragma unroll
    for (int j = 0; j < EPT / 4; ++j) {
        float4 o;
        o.x = x[4*j+0] - lse; o.y = x[4*j+1] - lse;
        o.z = x[4*j+2] - lse; o.w = x[4*j+3] - lse;
        lp4[j] = o;
    }
}

// ---- main: one block per row; K=32 relaxed top-1 softmax passes, state in VGPRs ----
__global__ void __launch_bounds__(TPB)
gumbel_topk_kernel(const float* __restrict__ u, const float* __restrict__ lp,
                   float* __restrict__ out) {
    __shared__ float s_red[NWAVES];
    const int t = threadIdx.x;
    const size_t row = blockIdx.x;

    float wcur[EPT], oh[EPT], acc[EPT];

    // r = log_proba + (-log(-log u))
    //   -log u      = log2(u)  * (-ln2)      (positive)
    //   -log(-logu) = log2(t)  * (-ln2)
    const float4* u4  = (const float4*)(u + row * (size_t)DIMN + t * EPT);
    const float4* lp4 = (const float4*)(lp + t * EPT);
#pragma unroll
    for (int j = 0; j < EPT / 4; ++j) {
        float4 uu = u4[j];
        float4 ll = lp4[j];
        float t0 = __builtin_amdgcn_logf(uu.x) * NLN2F;
        float t1 = __builtin_amdgcn_logf(uu.y) * NLN2F;
        float t2 = __builtin_amdgcn_logf(uu.z) * NLN2F;
        float t3 = __builtin_amdgcn_logf(uu.w) * NLN2F;
        wcur[4*j+0] = fmaf(__builtin_amdgcn_logf(t0), NLN2F, ll.x);
        wcur[4*j+1] = fmaf(__builtin_amdgcn_logf(t1), NLN2F, ll.y);
        wcur[4*j+2] = fmaf(__builtin_amdgcn_logf(t2), NLN2F, ll.z);
        wcur[4*j+3] = fmaf(__builtin_amdgcn_logf(t3), NLN2F, ll.w);
    }
#pragma unroll
    for (int i = 0; i < EPT; ++i) { oh[i] = 0.0f; acc[i] = 0.0f; }

    for (int k = 0; k < KITER; ++k) {
        // w_new = w_cur + log(max(1 - onehot, EPS)); track local max
        // log(mask) = v_log_f32(mask) * ln2, fused into the add via FMA
        float lm = NEGINF;
#pragma unroll
        for (int i = 0; i < EPT; ++i) {
            float mask = fmaxf(1.0f - oh[i], EPSF);
            wcur[i] = fmaf(__builtin_amdgcn_logf(mask), LN2F, wcur[i]);
            lm = fmaxf(lm, wcur[i]);
        }
        const float m = block_max(lm, s_red);

        // softmax(w_new / t): exp2((w_new - m) * 100*log2e), then normalize
        float ls = 0.0f;
#pragma unroll
        for (int i = 0; i < EPT; ++i) {
            float e = __builtin_amdgcn_exp2f((wcur[i] - m) * C100L2E);
            oh[i] = e;
            ls += e;
        }
        const float s = block_sum(ls, s_red);      // v_wmma path
        const float inv = __builtin_amdgcn_rcpf(s);
#pragma unroll
        for (int i = 0; i < EPT; ++i) {
            oh[i] *= inv;
            acc[i] += oh[i];
        }
    }

    float4* o4 = (float4*)(out + row * (size_t)DIMN + t * EPT);
#pragma unroll
    for (int j = 0; j < EPT / 4; ++j) {
        float4 o;
        o.x = acc[4*j+0]; o.y = acc[4*j+1];
        o.z = acc[4*j+2]; o.w = acc[4*j+3];
        o4[j] = o;
    }
}

extern "C" void kernel_launch(void* const* d_in, const int* in_sizes, int n_in,
                              void* d_out, int out_size, void* d_ws, size_t ws_size,
                              hipStream_t stream) {
    // setup_inputs order: inputs (dummy int), w (DIM f32), u (N*DIM f32)
    const float* w = (const float*)d_in[1];
    const float* u = (const float*)d_in[2];
    float* out = (float*)d_out;
    float* lp  = (float*)d_ws;               // DIM floats of scratch (16 KB)

    const int nrows = in_sizes[2] / DIMN;    // N = 2048

    gumbel_prep_logsoftmax<<<1, TPB, 0, stream>>>(w, lp);
    gumbel_topk_kernel<<<nrows, TPB, 0, stream>>>(u, lp, out);
}